// Loss_50500225466792
// MI455X (gfx1250) — compile-verified
//
#include <hip/hip_runtime.h>
#include <hip/hip_bf16.h>

typedef __attribute__((ext_vector_type(2))) float v2f;
typedef __attribute__((ext_vector_type(8))) float v8f;

#define SS 7
#define NB 2
#define NC 20
#define LL 49
#define PRED_STRIDE (LL * (NC + 5 * NB)) /* 1470 */
#define LAB_STRIDE  (LL * (1 + NC + 4))  /* 1225 */

__device__ __forceinline__ float nanz(float x) { return (x == x) ? x : 0.0f; }

__global__ void zero_out_kernel(float* out) {
    if (threadIdx.x == 0) out[0] = 0.0f;
}

__global__ __launch_bounds__(256) void yolo_loss_kernel(
    const float* __restrict__ preds,
    const float* __restrict__ labels,
    float* __restrict__ out,
    int total_cells)
{
    __shared__ float red[256];
    const int tid = threadIdx.x;
    const int idx = blockIdx.x * 256 + tid;

    float loss = 0.0f;

    if (idx < total_cells) {
        const int n = idx / LL;
        const int l = idx - n * LL;

        const float* pr = preds  + (size_t)n * PRED_STRIDE;
        const float* lb = labels + (size_t)n * LAB_STRIDE + l * (1 + NC + 4);

        // obj flag + both confidences (always needed).
        const float obj = lb[0];
        // pconf element offset = n*1470 + 980 + l*2  (even -> 8B aligned)
        const float2 conf = *reinterpret_cast<const float2*>(pr + LL * NC + l * NB);
        const float c0 = conf.x, c1 = conf.y;

        if (obj == 0.0f) {
            // noobj: conf_delta = NOOBJ * pconf^2 for both boxes; cls/coord are 0.
            loss = nanz(0.5f * c0 * c0) + nanz(0.5f * c1 * c1);
        } else {
            // ---- heavy path: load boxes (8B-aligned pairs) ----
            const float* pb = pr + LL * NC + LL * NB + l * (NB * 4);
            const float2 q0 = *reinterpret_cast<const float2*>(pb + 0);
            const float2 q1 = *reinterpret_cast<const float2*>(pb + 2);
            const float2 q2 = *reinterpret_cast<const float2*>(pb + 4);
            const float2 q3 = *reinterpret_cast<const float2*>(pb + 6);
            const float box[2][4] = {{q0.x, q0.y, q1.x, q1.y},
                                     {q2.x, q2.y, q3.x, q3.y}};

            const float tx = lb[1 + NC + 0];
            const float ty = lb[1 + NC + 1];
            const float tw = lb[1 + NC + 2];
            const float th = lb[1 + NC + 3];

            // t = [tbox.xy / S, tbox.wh]
            const float t0 = tx / 7.0f, t1 = ty / 7.0f, t2 = tw, t3 = th;

            float iou[2], rmse[2];
            #pragma unroll
            for (int b = 0; b < 2; ++b) {
                // o = [pbox.xy / S, pbox.wh ^ 2]  (stop_gradient irrelevant fwd)
                const float o0 = box[b][0] / 7.0f;
                const float o1 = box[b][1] / 7.0f;
                const float o2 = box[b][2] * box[b][2];
                const float o3 = box[b][3] * box[b][3];
                const float left   = fmaxf(t0 - 0.5f * t2, o0 - 0.5f * o2);
                const float right  = fminf(t0 + 0.5f * t2, o0 + 0.5f * o2);
                const float top    = fmaxf(t1 - 0.5f * t3, o1 - 0.5f * o3);
                const float bottom = fminf(t1 + 0.5f * t3, o1 + 0.5f * o3);
                const float w = right - left;
                const float h = bottom - top;
                const bool invalid = (w < 0.0f) || (h < 0.0f);
                const float inter = invalid ? 0.0f : w * h;
                const float uni = t2 * t3 + o2 * o3 - inter;
                iou[b] = invalid ? 0.0f : inter / fmaxf(uni, 1e-12f);
                const float d0 = t0 - o0, d1 = t1 - o1, d2 = t2 - o2, d3 = t3 - o3;
                rmse[b] = sqrtf(d0 * d0 + d1 * d1 + d2 * d2 + d3 * d3);
            }

            const float max_iou = fmaxf(iou[0], iou[1]);
            // jnp.argmax / argmin: first occurrence wins on ties.
            const int best = (max_iou > 0.0f) ? (iou[0] >= iou[1] ? 0 : 1)
                                              : (rmse[0] <= rmse[1] ? 0 : 1);
            const float best_iou  = best ? iou[1] : iou[0];
            const float conf_best = best ? c1 : c0;
            const float conf_oth  = best ? c0 : c1;

            // conf: OBJ term for best box, NOOBJ term for the other.
            const float e = best_iou - conf_best;
            loss += nanz(0.5f * e * e) + nanz(0.5f * conf_oth * conf_oth);

            // cls: CLS * (tcls - pcls)^2  (8B-aligned pair loads)
            const float* pcl = pr + l * NC;
            #pragma unroll
            for (int c = 0; c < NC; c += 2) {
                const float2 p2 = *reinterpret_cast<const float2*>(pcl + c);
                const float d0 = lb[1 + c] - p2.x;
                const float d1 = lb[2 + c] - p2.y;
                loss += nanz(0.5f * d0 * d0) + nanz(0.5f * d1 * d1);
            }

            // coord: COORD * (ttarget - pbox[best])^2, ttarget = [xy, sqrt(wh)]
            const float tt[4] = {tx, ty, sqrtf(tw), sqrtf(th)};
            #pragma unroll
            for (int j = 0; j < 4; ++j) {
                const float d = tt[j] - box[best][j];
                loss += nanz(2.5f * d * d);
            }
        }
    }

    // ---- block reduction: LDS -> wave0 WMMA (A = ones) -> shuffle -> atomic ----
    red[tid] = loss;
    __syncthreads();

    if (tid < 32) {
        // A = all-ones 16x4 (2 VGPRs x 32 lanes = 64 slots, bijective onto A).
        v2f a; a.x = 1.0f; a.y = 1.0f;
        v8f acc = {};
        #pragma unroll
        for (int chunk = 0; chunk < 4; ++chunk) {
            // B = 64 partials per chunk; any bijection onto the 4x16 matrix is
            // fine since A=1 makes D's rows all equal the column sums.
            v2f b;
            b.x = red[chunk * 64 + tid];
            b.y = red[chunk * 64 + 32 + tid];
            acc = __builtin_amdgcn_wmma_f32_16x16x4_f32(
                false, a, false, b, (short)0, acc, false, false);
        }
        // D VGPR0 holds two complete rows (M=0 and M=8); each row is the full
        // set of column sums, so summing VGPR0 over 32 lanes gives 2 * total.
        float s = acc[0];
        #pragma unroll
        for (int off = 16; off > 0; off >>= 1)
            s += __shfl_xor(s, off, 32);
        if (tid == 0) atomicAdd(out, 0.5f * s);
    }
}

extern "C" void kernel_launch(void* const* d_in, const int* in_sizes, int n_in,
                              void* d_out, int out_size, void* d_ws, size_t ws_size,
                              hipStream_t stream) {
    const float* preds  = (const float*)d_in[0];
    const float* labels = (const float*)d_in[1];
    float* out = (float*)d_out;

    const int n = in_sizes[0] / PRED_STRIDE;   // 16384
    const int total_cells = n * LL;            // 802816
    const int blocks = (total_cells + 255) / 256;

    zero_out_kernel<<<1, 32, 0, stream>>>(out);
    yolo_loss_kernel<<<blocks, 256, 0, stream>>>(preds, labels, out, total_cells);
}